// GARCHGenerator_73237782331636
// MI455X (gfx1250) — compile-verified
//
#include <hip/hip_runtime.h>
#include <cstdint>

// GARCH(1,1) path generation, parallelized via contraction warm-up chunking.
//
//   N = 24*131072 = 3,145,728 steps/series, 3 series. Chunks of L=512 outputs
//   with W=512 warm-up steps (contraction E[alpha*z^2+beta] < ~0.96 =>
//   initial-state error < e^-25, below fp32 ulp). 18432 independent lanes,
//   1023 steps each, ~2x work inflation vs. the (unusable) sequential scan.
//
// CDNA5 data path: one wave per block; its 32 lanes own 32 consecutive chunks.
// Noise staged in 8 "supertiles" of 128 slots/owner via
// global_load_async_to_lds_b128 (one instruction per owner row = 512B fully
// coalesced, DMA direct to LDS, tracked by ASYNCcnt), double-buffered with
// s_wait_asynccnt. Slots start at t0 (even) so every global address is 16B
// aligned and no upper clamp is needed. Outputs re-phased to start at t=iL so
// they batch into 128 global_store_b128 per lane (16B aligned).

#define N_TOT   3145728                 // 24 * 131072
#define LCH     512                     // outputs per chunk
#define WUP     512                     // warm-up steps
#define NCH     (N_TOT / LCH)           // 6144 chunks per series
#define NTASK   (3 * NCH)               // 18432 lanes total
#define STILE   128                     // slots per supertile per owner
#define ROWD    132                     // LDS row stride in dwords (16B aligned)

__global__ __launch_bounds__(32)
void garch_scan_kernel(const float* __restrict__ params,
                       const float* __restrict__ noise,
                       float* __restrict__ out)
{
    // [buf][owner][slot]; 2*32*132*4 = 33792 B, one wave per block.
    __shared__ __attribute__((aligned(16))) float sh[2][32][ROWD];

    const int lane      = threadIdx.x & 31;
    const int task      = blockIdx.x * 32 + threadIdx.x;       // < NTASK
    const int series    = task / NCH;                          // wave-uniform
    const int chunk     = task % NCH;
    const int chunkBase = chunk - lane;                        // wave's first chunk

    // Only the first wave of series 0 reads before t=0 (clamped, whole 16B
    // chunks: the negative region is exactly slots 0..511, a multiple of 4).
    const bool needClamp = (series == 0 && chunkBase == 0);

    const float mu    = params[series * 4 + 0];
    const float omega = params[series * 4 + 1];
    const float alpha = params[series * 4 + 2];
    const float beta  = params[series * 4 + 3];

    // Warm-up start: stationary variance (fastest convergence).
    float v = omega / fmaxf(1.0f - alpha - beta, 1e-3f);
    float p = mu;

    const uint64_t noiseAddr = (uint64_t)(uintptr_t)noise;
    float* const   outBase   = out + (long)series * N_TOT;

    // Stage supertile st into buffer bufIdx: for each owner o, lane j DMAs
    // 16B (slots 4j..4j+3) -> 512B coalesced per instruction.
    auto stage = [&](int st, int bufIdx) {
        const unsigned ldsBase =
            (unsigned)(uintptr_t)(&sh[bufIdx][0][0]) + (unsigned)lane * 16u;
        const long idx0 = (long)series * N_TOT + (long)chunkBase * LCH
                          - WUP + (long)st * STILE + lane * 4;   // multiple of 4
        if (needClamp) {
            #pragma unroll 4
            for (int o = 0; o < 32; ++o) {
                long idx = idx0 + (long)o * LCH;
                idx = idx < 0 ? 0 : idx;                         // stays 16B aligned
                uint64_t g = noiseAddr + ((uint64_t)idx << 2);
                unsigned  l = ldsBase + (unsigned)(o * (ROWD * 4));
                asm volatile("global_load_async_to_lds_b128 %0, %1, off"
                             :: "v"(l), "v"(g) : "memory");
            }
        } else {
            uint64_t g = noiseAddr + ((uint64_t)idx0 << 2);
            unsigned  l = ldsBase;
            #pragma unroll 4
            for (int o = 0; o < 32; ++o) {
                asm volatile("global_load_async_to_lds_b128 %0, %1, off"
                             :: "v"(l), "v"(g) : "memory");
                g += (uint64_t)(LCH * 4);
                l += ROWD * 4;
            }
        }
    };

    #define GARCH_STEP(zval)                                          \
        do {                                                          \
            v = fmaf(beta, v, fmaf(alpha, p * p, omega));             \
            p = fmaf((zval), __builtin_amdgcn_sqrtf(v), mu);          \
        } while (0)

    int tq = chunk * LCH;   // next output index (advances in groups of 4)

    // Warm-up consume: slots k0..127 of a buffer, no writes.
    auto consumeWarm = [&](int bufIdx, int k0) {
        const float* row = &sh[bufIdx][lane][0];
        #pragma unroll 8
        for (int k = k0; k < STILE; ++k) { GARCH_STEP(row[k]); }
    };

    // Write consume: 128 slots, batched into 32 float4 stores.
    auto consumeWrite = [&](int bufIdx) {
        const float* row = &sh[bufIdx][lane][0];
        #pragma unroll 4
        for (int kk = 0; kk < STILE; kk += 4) {
            float4 acc;
            GARCH_STEP(row[kk + 0]); acc.x = p;
            GARCH_STEP(row[kk + 1]); acc.y = p;
            GARCH_STEP(row[kk + 2]); acc.z = p;
            GARCH_STEP(row[kk + 3]); acc.w = p;
            *(float4*)(outBase + tq) = acc; tq += 4;
        }
    };

    // ---- Pipeline: 8 supertiles, double-buffered (st & 1 selects buffer) ----
    stage(0, 0);
    stage(1, 1);
    asm volatile("s_wait_asynccnt 32" ::: "memory");   // st0 resident
    consumeWarm(0, 1);                                 // slots 1..127  (127 warm)

    stage(2, 0);
    asm volatile("s_wait_asynccnt 32" ::: "memory");
    consumeWarm(1, 0);                                 // 128 warm

    stage(3, 1);
    asm volatile("s_wait_asynccnt 32" ::: "memory");
    consumeWarm(0, 0);                                 // 128 warm

    stage(4, 0);
    asm volatile("s_wait_asynccnt 32" ::: "memory");
    consumeWarm(1, 0);                                 // 128 warm (511 total)

    stage(5, 1);
    asm volatile("s_wait_asynccnt 32" ::: "memory");
    {   // st4: one warm step (t = iL), boundary, then 127 written steps.
        const float* row = &sh[0][lane][0];
        GARCH_STEP(row[0]);                            // state -> (p_iL, v_iL)
        if (chunk == 0) { p = 0.0f; v = 0.0f; }        // exact start at t=0
        float4 acc; acc.x = p;                         // out[iL]
        GARCH_STEP(row[1]); acc.y = p;
        GARCH_STEP(row[2]); acc.z = p;
        GARCH_STEP(row[3]); acc.w = p;
        *(float4*)(outBase + tq) = acc; tq += 4;
        #pragma unroll 4
        for (int kk = 4; kk < STILE; kk += 4) {
            float4 a2;
            GARCH_STEP(row[kk + 0]); a2.x = p;
            GARCH_STEP(row[kk + 1]); a2.y = p;
            GARCH_STEP(row[kk + 2]); a2.z = p;
            GARCH_STEP(row[kk + 3]); a2.w = p;
            *(float4*)(outBase + tq) = a2; tq += 4;
        }
    }

    stage(6, 0);
    asm volatile("s_wait_asynccnt 32" ::: "memory");
    consumeWrite(1);                                   // t = iL+128 .. iL+255

    stage(7, 1);
    asm volatile("s_wait_asynccnt 32" ::: "memory");
    consumeWrite(0);                                   // t = iL+256 .. iL+383

    asm volatile("s_wait_asynccnt 0" ::: "memory");
    consumeWrite(1);                                   // t = iL+384 .. iL+511

    #undef GARCH_STEP
}

extern "C" void kernel_launch(void* const* d_in, const int* in_sizes, int n_in,
                              void* d_out, int out_size, void* d_ws, size_t ws_size,
                              hipStream_t stream)
{
    const float* params = (const float*)d_in[0];  // (3,4): mu, omega, alpha, beta
    const float* noise  = (const float*)d_in[1];  // (3, N) float32
    float* out = (float*)d_out;                   // 3*N floats

    dim3 grid(NTASK / 32);   // 576 blocks
    dim3 block(32);          // one wave; owns 32 consecutive chunks
    hipLaunchKernelGGL(garch_scan_kernel, grid, block, 0, stream,
                       params, noise, out);
}